// GraphConvLSTM_47802986005060
// MI455X (gfx1250) — compile-verified
//
#include <hip/hip_runtime.h>
#include <stdint.h>
#include <math.h>

#define SEQ_LEN   100
#define N_NODES   22
#define HID       3
#define IN_DIM    32
#define BATCH_TOT 16384
#define OUT_STRIDE (SEQ_LEN * N_NODES * HID)   // 6600 floats per batch row

#define WAVES_PER_BLOCK 4
#define BPW             16          // batch elements per wave (WMMA N)
#define SLOT            66          // floats per step per batch (22 nodes * 3)
#define ROWF            (2 * SLOT)  // 2-step staging row per batch = 132 floats

typedef __attribute__((ext_vector_type(2))) float v2f;
typedef __attribute__((ext_vector_type(8))) float v8f;

__device__ __forceinline__ float fast_tanh(float x) {
#if __has_builtin(__builtin_amdgcn_tanhf)
    return __builtin_amdgcn_tanhf(x);    // v_tanh_f32 on gfx1250
#else
    return tanhf(x);
#endif
}

// ---- constant normalized adjacency, synthesized from clique bitmasks ----
__device__ __forceinline__ uint32_t node_mask(int i) {
    const uint32_t S0 = 0x00000925u;  // {0,2,5,8,11}
    const uint32_t S1 = 0x00000493u;  // {0,1,4,7,10}
    const uint32_t S2 = 0x00009249u;  // {0,3,6,9,12,15}
    const uint32_t S3 = 0x002A4200u;  // {9,14,17,19,21}
    const uint32_t S4 = 0x00152200u;  // {9,13,16,18,20}
    uint32_t bit = 1u << i, m = 0u;
    if (S0 & bit) m |= S0;
    if (S1 & bit) m |= S1;
    if (S2 & bit) m |= S2;
    if (S3 & bit) m |= S3;
    if (S4 & bit) m |= S4;
    return m & ~bit;
}

__device__ __forceinline__ float ahat_val(int i, int j) {
    if (i >= N_NODES || j >= N_NODES) return 0.0f;
    uint32_t mi = node_mask(i);
    if (!((mi >> j) & 1u)) return 0.0f;
    int di = __popc(mi);
    int dj = __popc(node_mask(j));
    return rsqrtf((float)(di * dj));    // D^{-1/2} A D^{-1/2}
}

__global__ __launch_bounds__(WAVES_PER_BLOCK * 32)
void gclstm_kernel(const float* __restrict__ x,
                   const float* __restrict__ Wi, const float* __restrict__ bi,
                   const float* __restrict__ Wf, const float* __restrict__ bf,
                   const float* __restrict__ Wo, const float* __restrict__ bo,
                   const float* __restrict__ Wc, const float* __restrict__ bc,
                   const float* __restrict__ Wg, const float* __restrict__ bg,
                   float* __restrict__ out)
{
    // per-wave staging: [batch 16][2 steps * 66 floats], b-major so a 2-step
    // flush is 132 contiguous floats (= 33 float4) per batch element.
    __shared__ __align__(16) float stage[WAVES_PER_BLOCK][BPW][ROWF];

    const int tid  = threadIdx.x;
    const int wave = tid >> 5;
    const int lane = tid & 31;
    const int bl   = lane & 15;     // WMMA N / local batch
    const int hi   = lane >> 4;     // upper lane half
    const int gb0  = (blockIdx.x * WAVES_PER_BLOCK + wave) * BPW;
    const int gb   = gb0 + bl;

    // ---- zero this wave's staging region (slot for h_{-1}=0 + pad safety) ----
    float* wstage = &stage[wave][0][0];
    for (int i = lane; i < BPW * ROWF; i += 32) wstage[i] = 0.0f;

    // ---- A-hat WMMA A-fragments: frag[mt][kc] -> rows mt*16+bl, K=kc*4+v+2*hi ----
    float af[2][6][2];
#pragma unroll
    for (int mt = 0; mt < 2; ++mt)
#pragma unroll
        for (int kc = 0; kc < 6; ++kc)
#pragma unroll
            for (int v = 0; v < 2; ++v)
                af[mt][kc][v] = ahat_val(mt * 16 + bl, kc * 4 + v + 2 * hi);

    // ---- per-lane B-fragment node offsets (clamp K>=22: A column is zero) ----
    int noff[6][2];
#pragma unroll
    for (int kc = 0; kc < 6; ++kc)
#pragma unroll
        for (int v = 0; v < 2; ++v) {
            int node = 4 * kc + v + 2 * hi;
            if (node > 21) node = 0;           // multiplied by af==0 anyway
            noff[kc][v] = node * 3;
        }

    // ---- small weights ----
    float wg[3][3], bgv[3];
#pragma unroll
    for (int k = 0; k < 3; ++k) {
        bgv[k] = bg[k];
#pragma unroll
        for (int j = 0; j < 3; ++j) wg[k][j] = Wg[k * 3 + j];
    }

    // ---- timestep-invariant gate projections (pre-scaled for tanh-sigmoid) ----
    float xih[3], xfh[3], xoh[3], xcf[3];
    {
        float xr[IN_DIM];
#pragma unroll
        for (int c = 0; c < IN_DIM; c += 4) {
            float4 v = *(const float4*)(x + (size_t)gb * IN_DIM + c);
            xr[c] = v.x; xr[c + 1] = v.y; xr[c + 2] = v.z; xr[c + 3] = v.w;
        }
#pragma unroll
        for (int k = 0; k < 3; ++k) {
            float si = bi[k], sf = bf[k], so = bo[k], sc = bc[k];
#pragma unroll
            for (int c = 0; c < IN_DIM; ++c) {
                si = fmaf(xr[c], Wi[k * IN_DIM + c], si);
                sf = fmaf(xr[c], Wf[k * IN_DIM + c], sf);
                so = fmaf(xr[c], Wo[k * IN_DIM + c], so);
                sc = fmaf(xr[c], Wc[k * IN_DIM + c], sc);
            }
            xih[k] = 0.5f * si; xfh[k] = 0.5f * sf; xoh[k] = 0.5f * so;
            xcf[k] = sc;
        }
    }

    // ---- cell state in registers (valid rows only: 8 + 6 node-rows) ----
    float c0[8][3], c1[6][3];
#pragma unroll
    for (int r = 0; r < 8; ++r)
#pragma unroll
        for (int k = 0; k < 3; ++k) c0[r][k] = 0.0f;
#pragma unroll
    for (int r = 0; r < 6; ++r)
#pragma unroll
        for (int k = 0; k < 3; ++k) c1[r][k] = 0.0f;

    float* myrow = &stage[wave][bl][0];

#pragma unroll 2
    for (int t = 0; t < SEQ_LEN; ++t) {
        const int scur = t & 1;
        const int sprv = scur ^ 1;

        // ---- gather B fragments (h_{t-1}) from LDS staging ----
        float bfr[6][3][2];
        {
            const float* rp = myrow + sprv * SLOT;
#pragma unroll
            for (int kc = 0; kc < 6; ++kc)
#pragma unroll
                for (int v = 0; v < 2; ++v)
#pragma unroll
                    for (int j = 0; j < 3; ++j)
                        bfr[kc][j][v] = rp[noff[kc][v] + j];
        }

        // ---- tmp = A_hat @ h  via  v_wmma_f32_16x16x4_f32 ----
        v8f acc[3][2];
#pragma unroll
        for (int j = 0; j < 3; ++j)
#pragma unroll
            for (int mt = 0; mt < 2; ++mt) {
                v8f a = {0.f, 0.f, 0.f, 0.f, 0.f, 0.f, 0.f, 0.f};
#pragma unroll
                for (int kc = 0; kc < 6; ++kc) {
                    v2f av; av[0] = af[mt][kc][0]; av[1] = af[mt][kc][1];
                    v2f bv; bv[0] = bfr[kc][j][0]; bv[1] = bfr[kc][j][1];
                    a = __builtin_amdgcn_wmma_f32_16x16x4_f32(
                            false, av, false, bv, (short)0, a, false, false);
                }
                acc[j][mt] = a;
            }

        // ---- Wg contraction + gates + LSTM update + stage h_t ----
        float* wr = myrow + scur * SLOT;
#pragma unroll
        for (int mt = 0; mt < 2; ++mt) {
            const int RMAX = mt ? 6 : 8;
#pragma unroll
            for (int r = 0; r < RMAX; ++r) {
                float t0 = acc[0][mt][r];
                float t1 = acc[1][mt][r];
                float t2 = acc[2][mt][r];
#pragma unroll
                for (int k = 0; k < 3; ++k) {
                    float hc  = fmaf(t0, wg[k][0],
                                fmaf(t1, wg[k][1],
                                fmaf(t2, wg[k][2], bgv[k])));
                    float hch = 0.5f * hc;
                    float ig = fmaf(0.5f, fast_tanh(xih[k] + hch), 0.5f);
                    float fg = fmaf(0.5f, fast_tanh(xfh[k] + hch), 0.5f);
                    float og = fmaf(0.5f, fast_tanh(xoh[k] + hch), 0.5f);
                    float ct = fast_tanh(xcf[k] + hc);
                    float cp = mt ? c1[r][k] : c0[r][k];
                    float cn = fmaf(fg, cp, ig * ct);
                    if (mt) c1[r][k] = cn; else c0[r][k] = cn;
                    float hn = og * fast_tanh(cn);
                    if (mt == 0) {
                        wr[(r + 8 * hi) * 3 + k] = hn;     // nodes 0..15
                    } else if (!hi) {
                        wr[(16 + r) * 3 + k] = hn;         // nodes 16..21
                    }
                }
            }
        }

        // ---- flush 2 buffered steps: coalesced float4 global stores ----
        if (scur == 1) {
            const float4* src = (const float4*)wstage;   // [b][33] float4, b-major
            const size_t tofs = (size_t)(t - 1) * SLOT;
            for (int id = lane; id < BPW * 33; id += 32) {
                int b = id / 33;
                int c = id - b * 33;
                float4 v = src[id];
                *(float4*)(out + (size_t)(gb0 + b) * OUT_STRIDE + tofs + (size_t)c * 4) = v;
            }
        }
    }
}

extern "C" void kernel_launch(void* const* d_in, const int* in_sizes, int n_in,
                              void* d_out, int out_size, void* d_ws, size_t ws_size,
                              hipStream_t stream) {
    (void)in_sizes; (void)n_in; (void)d_ws; (void)ws_size; (void)out_size;
    const float* x  = (const float*)d_in[0];
    const float* Wi = (const float*)d_in[1];
    const float* bi = (const float*)d_in[2];
    const float* Wf = (const float*)d_in[3];
    const float* bf = (const float*)d_in[4];
    const float* Wo = (const float*)d_in[5];
    const float* bo = (const float*)d_in[6];
    const float* Wc = (const float*)d_in[7];
    const float* bc = (const float*)d_in[8];
    const float* Wg = (const float*)d_in[9];
    const float* bg = (const float*)d_in[10];
    float* out = (float*)d_out;

    const int blocks = BATCH_TOT / (WAVES_PER_BLOCK * BPW);   // 256
    gclstm_kernel<<<blocks, WAVES_PER_BLOCK * 32, 0, stream>>>(
        x, Wi, bi, Wf, bf, Wo, bo, Wc, bc, Wg, bg, out);
}